// ManeuverPrediction_4217657885154
// MI455X (gfx1250) — compile-verified
//
#include <hip/hip_runtime.h>

// ---------------------------------------------------------------------------
// ManeuverPrediction forward for gfx1250 (MI455X).
// All GEMMs (GAT projections, LSTM recurrences, decoder MLPs) run on the f32
// matrix pipe via V_WMMA_F32_16X16X4_F32.  LSTM recurrences are persistent
// kernels: h lives in LDS, c lives in accumulator VGPRs, one wave owns the
// 4 gate tiles of each 16-wide hidden tile so gate math is fragment-local.
// Hot loops are branchless: OOB handling is clamp-address + mask-value, and
// the N-tail mask path is selected by a *scalar* (blockIdx-based) branch so
// the dominant GEMMs run a zero-overhead unmasked inner loop.
//
// Input order assumption: d_in = tree_leaves(setup_inputs()) with dicts
// iterated in sorted-key order (jax convention):
//   0 adj(i32) 1 node_mask(i32) 2 nodes
//   3 b_dec 4 b_dev 5 b_geo 6 b_surr 7 b_traj
//   8..10  bilstm1_b (Wx,Wh,b)   11..13 bilstm1_f
//   14..16 bilstm2_b             17..19 bilstm2_f
//   20..28 dec_lat (Wd0,Wd1,Wo,Wout,Wv,bd0,bd1,bo,bout)
//   29..37 dec_lon (same order)
//   38..40 emb_dev  41..43 emb_geo  44..46 emb_surr  47..49 emb_traj
//   50 g_dec 51 g_dev 52 g_geo 53 g_surr 54 g_traj
//   55 gat1_W 56 gat1_ad 57 gat1_as 58 gat2_W 59 gat2_ad 60 gat2_as
//   61 x_lane_dev 62 x_lane_geo 63 x_traj
// Output: d_out[0:768] = softmax lat (256,3), d_out[768:1792] = lon (256,4).
// ---------------------------------------------------------------------------

typedef float v2f __attribute__((ext_vector_type(2)));
typedef float v8f __attribute__((ext_vector_type(8)));

#define TT 50

__device__ __forceinline__ v8f wmma_f32_k4(v8f c, float a0, float a1, float b0, float b1) {
  v2f a; a[0] = a0; a[1] = a1;
  v2f b; b[0] = b0; b[1] = b1;
  // D = A(16x4) * B(4x16) + C, f32, wave32
  return __builtin_amdgcn_wmma_f32_16x16x4_f32(false, a, false, b, (short)0, c, false, false);
}

__device__ __forceinline__ float sigf(float x) { return 1.f / (1.f + __expf(-x)); }
__device__ __forceinline__ int   imin(int a, int b) { return a < b ? a : b; }

// ---------------------------------------------------------------------------
// Generic f32 WMMA GEMM: C[M,N] = act(A[M,K] @ B[K,N] + bias).  K % 4 == 0.
// Block: 256 threads (8 waves). Tile: 16 x 128 (one 16x16 per wave).
// A staged through LDS (row pad 66 -> conflict-free b64 ds reads).
// N tail: scalar (SGPR) branch selects masked path; full blocks (all large
// GEMMs have N % 128 == 0) run unmasked loads straight into fragments.
// act: 0 = none, 1 = leaky_relu(0.2)
// ---------------------------------------------------------------------------
__global__ void gemm_f32_wmma(const float* __restrict__ A, const float* __restrict__ B,
                              const float* __restrict__ bias, float* __restrict__ C,
                              int M, int N, int K, int lda, int ldb, int ldc, int act)
{
  __shared__ float sA[16][66];
  const int tid  = threadIdx.x;
  const int wave = tid >> 5;
  const int lane = tid & 31;
  const int lm   = lane & 15;
  const int kh   = lane >> 4;               // 0: K rows {0,1}; 1: {2,3}
  const int m0   = blockIdx.y << 4;
  const int n0   = (blockIdx.x << 7) + (wave << 4);
  const int col  = n0 + lm;
  const bool fullN = (((int)blockIdx.x + 1) << 7) <= N;   // scalar condition
  const int colc = imin(col, N - 1);        // clamped (always a valid address)
  const float nm = (col < N) ? 1.f : 0.f;   // value mask for N tail

  v8f acc;
#pragma unroll
  for (int r = 0; r < 8; ++r) acc[r] = 0.f;

  for (int kb = 0; kb < K; kb += 64) {
    // ---- stage A[16][64] into LDS (float4 fast path, clamped tail) ----
    {
      const int r    = tid >> 4;
      const int c0   = (tid & 15) << 2;
      const int krow = m0 + r;
      if (krow < M && (kb + c0 + 3) < K) {
        const float4 v = *(const float4*)(A + (long)krow * lda + kb + c0);
        sA[r][c0 + 0] = v.x; sA[r][c0 + 1] = v.y;
        sA[r][c0 + 2] = v.z; sA[r][c0 + 3] = v.w;
      } else {
        const long ar = (long)imin(krow, M - 1) * lda;
#pragma unroll
        for (int i = 0; i < 4; ++i) {
          const int k = kb + c0 + i;
          const float mv = (krow < M && k < K) ? 1.f : 0.f;
          sA[r][c0 + i] = A[ar + imin(k, K - 1)] * mv;
        }
      }
    }
    __syncthreads();

    const int kend = (K - kb) < 64 ? (K - kb) : 64;   // multiple of 4
    if (kb + 64 < K) __builtin_prefetch(B + (long)(kb + 64 + kh * 2) * ldb + colc, 0, 1);
    if (fullN) {
      // zero-overhead path: whole 128-wide block tile is in range
      const float* Bp = B + (long)(kb + kh * 2) * ldb + col;
#pragma unroll 4
      for (int kk = 0; kk < kend; kk += 4) {
        const float a0 = sA[lm][kk + kh * 2];
        const float a1 = sA[lm][kk + kh * 2 + 1];
        const float b0 = Bp[0];
        const float b1 = Bp[ldb];
        acc = wmma_f32_k4(acc, a0, a1, b0, b1);
        Bp += (long)4 * ldb;
      }
    } else {
      const float* Bp = B + (long)(kb + kh * 2) * ldb + colc;
#pragma unroll 4
      for (int kk = 0; kk < kend; kk += 4) {
        const float a0 = sA[lm][kk + kh * 2];
        const float a1 = sA[lm][kk + kh * 2 + 1];
        const float b0 = Bp[0]   * nm;
        const float b1 = Bp[ldb] * nm;
        acc = wmma_f32_k4(acc, a0, a1, b0, b1);
        Bp += (long)4 * ldb;
      }
    }
    __syncthreads();
  }

  const float bv = (bias != nullptr) ? (fullN ? bias[col] : bias[colc] * nm) : 0.f;
#pragma unroll
  for (int r = 0; r < 8; ++r) {
    const int row = m0 + r + (kh << 3);
    float v = acc[r] + bv;
    if (act == 1) v = v > 0.f ? v : 0.2f * v;
    if (row < M && col < N) C[(long)row * ldc + col] = v;
  }
}

// ---------------------------------------------------------------------------
// Fused BN + LSTM, H = 64.  Block: 128 threads (4 waves), 16 sequences/block.
// Wave w owns hidden tile w (16 cols) -> computes its i/f/g/o gate tiles.
// x@Wx fused per step (K padded to 8, clamp+mask); h@Wh from LDS h (K=64).
// Sequence s -> (b = s/npb, node = noff + s%npb); x indexed via inN nodes.
// Output element: out[((b*outN + node)*TT + t)*ostride + ocol + hid].
// ---------------------------------------------------------------------------
__global__ void emb_lstm64(const float* __restrict__ x, const float* __restrict__ gT,
                           const float* __restrict__ bT,
                           const float* __restrict__ Wx, const float* __restrict__ Wh,
                           const float* __restrict__ bz, float* __restrict__ out,
                           int nseq, int inN, int npb, int noff, int F,
                           int outN, int ostride, int ocol)
{
  __shared__ float hsh[16][68];               // row pad 68 -> conflict-free
  const int tid  = threadIdx.x;
  const int wave = tid >> 5;                  // j-tile 0..3
  const int lane = tid & 31;
  const int lm   = lane & 15;
  const int kh   = lane >> 4;
  const int rows0 = blockIdx.x << 4;

  const int   sA   = imin(rows0 + lm, nseq - 1);     // clamped lane row
  const float rms  = (rows0 + lm) < nseq ? 1.f : 0.f;
  const int   bA   = sA / npb;
  const int   nA   = noff + sA % npb;
  const long  xrow = ((long)(bA * inN + nA)) * TT * F;

  const int colI = (wave << 4) + lm;          // hidden col 0..63

  for (int i = tid; i < 16 * 68; i += blockDim.x) (&hsh[0][0])[i] = 0.f;
  v8f cs;
#pragma unroll
  for (int r = 0; r < 8; ++r) cs[r] = 0.f;
  __syncthreads();

  const float bi = bz[colI], bf = bz[64 + colI], bg = bz[128 + colI], bo = bz[192 + colI];

  for (int t = 0; t < TT; ++t) {
    const float gt = gT[t], bt = bT[t];
    v8f zi, zf, zg, zo;
#pragma unroll
    for (int r = 0; r < 8; ++r) { zi[r] = bi; zf[r] = bf; zg[r] = bg; zo[r] = bo; }

    // x_t @ Wx  (K padded to 8; clamp addresses, mask values -> branchless)
#pragma unroll
    for (int k0 = 0; k0 < 8; k0 += 4) {
      const int   ka  = k0 + kh * 2;
      const float m0v = (ka     < F) ? rms : 0.f;
      const float m1v = ((ka+1) < F) ? rms : 0.f;
      const int   k0c = imin(ka, F - 1);
      const int   k1c = imin(ka + 1, F - 1);
      const float a0 = (x[xrow + (long)t * F + k0c] * gt + bt) * m0v;
      const float a1 = (x[xrow + (long)t * F + k1c] * gt + bt) * m1v;
#pragma unroll
      for (int G = 0; G < 4; ++G) {
        const int c = (G << 6) + colI;
        const float b0 = Wx[k0c * 256 + c] * ((ka     < F) ? 1.f : 0.f);
        const float b1 = Wx[k1c * 256 + c] * (((ka+1) < F) ? 1.f : 0.f);
        v8f& z = G == 0 ? zi : (G == 1 ? zf : (G == 2 ? zg : zo));
        z = wmma_f32_k4(z, a0, a1, b0, b1);
      }
    }
    // h_{t-1} @ Wh  (K = 64, no guards needed)
#pragma unroll 2
    for (int kt = 0; kt < 64; kt += 4) {
      const int ka = kt + kh * 2;
      const float a0 = hsh[lm][ka];
      const float a1 = hsh[lm][ka + 1];
#pragma unroll
      for (int G = 0; G < 4; ++G) {
        const int c = (G << 6) + colI;
        const float b0 = Wh[ka * 256 + c];
        const float b1 = Wh[(ka + 1) * 256 + c];
        v8f& z = G == 0 ? zi : (G == 1 ? zf : (G == 2 ? zg : zo));
        z = wmma_f32_k4(z, a0, a1, b0, b1);
      }
    }
    __syncthreads();
#pragma unroll
    for (int r = 0; r < 8; ++r) {
      const float iv = sigf(zi[r]), fv = sigf(zf[r]);
      const float gv = tanhf(zg[r]), ov = sigf(zo[r]);
      const float cv = fv * cs[r] + iv * gv;
      cs[r] = cv;
      const float hv = ov * tanhf(cv);
      const int row = r + (kh << 3);
      hsh[row][colI] = hv;
      const int s = rows0 + row;
      if (s < nseq) {
        const int b = s / npb, nn = noff + s % npb;
        out[((long)(b * outN + nn) * TT + t) * ostride + ocol + colI] = hv;
      }
    }
    __syncthreads();
  }
}

// ---------------------------------------------------------------------------
// LSTM, H = 256, input projection precomputed: z = xp[b,t,:] + h @ Wh.
// Block: 256 threads (8 waves), 16 batch rows/block. Wave w owns hidden
// tiles {w, w+8}; per tile the 4 gate tiles live in its own fragments.
// dir=+1 forward, -1 backward (reads/writes time index TT-1-t).
// Output element: out[row*obstride + tx*otstride + ocol + hid].
// ---------------------------------------------------------------------------
__global__ void big_lstm256(const float* __restrict__ xp, const float* __restrict__ Wh,
                            float* __restrict__ out, int dir,
                            long obstride, int otstride, int ocol)
{
  __shared__ float hsh[16][260];              // 256 + pad -> conflict-free
  const int tid  = threadIdx.x;
  const int wave = tid >> 5;
  const int lane = tid & 31;
  const int lm   = lane & 15;
  const int kh   = lane >> 4;
  const int rows0 = blockIdx.x << 4;

  for (int i = tid; i < 16 * 260; i += blockDim.x) (&hsh[0][0])[i] = 0.f;
  v8f cs[2];
#pragma unroll
  for (int j = 0; j < 2; ++j)
#pragma unroll
    for (int r = 0; r < 8; ++r) cs[j][r] = 0.f;
  __syncthreads();

  for (int t = 0; t < TT; ++t) {
    const int tx = dir > 0 ? t : (TT - 1 - t);
    v8f z[2][4];
#pragma unroll
    for (int jj = 0; jj < 2; ++jj) {
      const int jc = ((wave + jj * 8) << 4) + lm;
#pragma unroll
      for (int G = 0; G < 4; ++G) {
        const int c = (G << 8) + jc;
#pragma unroll
        for (int r = 0; r < 8; ++r) {
          const int row = rows0 + r + (kh << 3);
          z[jj][G][r] = xp[((long)row * TT + tx) * 1024 + c];
        }
      }
    }
#pragma unroll 2
    for (int kt = 0; kt < 256; kt += 4) {
      const int ka = kt + kh * 2;
      const float a0 = hsh[lm][ka];
      const float a1 = hsh[lm][ka + 1];
#pragma unroll
      for (int jj = 0; jj < 2; ++jj) {
        const int jc = ((wave + jj * 8) << 4) + lm;
#pragma unroll
        for (int G = 0; G < 4; ++G) {
          const int c = (G << 8) + jc;
          const float b0 = Wh[(long)ka * 1024 + c];
          const float b1 = Wh[(long)(ka + 1) * 1024 + c];
          z[jj][G] = wmma_f32_k4(z[jj][G], a0, a1, b0, b1);
        }
      }
    }
    __syncthreads();
#pragma unroll
    for (int jj = 0; jj < 2; ++jj) {
      const int jc = ((wave + jj * 8) << 4) + lm;
#pragma unroll
      for (int r = 0; r < 8; ++r) {
        const float iv = sigf(z[jj][0][r]);
        const float fv = sigf(z[jj][1][r]);
        const float gv = tanhf(z[jj][2][r]);
        const float ov = sigf(z[jj][3][r]);
        const float cv = fv * cs[jj][r] + iv * gv;
        cs[jj][r] = cv;
        const float hv = ov * tanhf(cv);
        const int row = r + (kh << 3);
        hsh[row][jc] = hv;
        out[(long)(rows0 + row) * obstride + (long)tx * otstride + ocol + jc] = hv;
      }
    }
    __syncthreads();
  }
}

// ---------------------------------------------------------------------------
// GAT attention logits: es/ed[b,h,n] = <Wh[b,n,h,:], a_src/dst[h,:]>
// One block per (b,n); one wave per head; wave32 shuffle reduction.
// ---------------------------------------------------------------------------
__global__ void gat_esed(const float* __restrict__ wh, const float* __restrict__ asv,
                         const float* __restrict__ adv,
                         float* __restrict__ es, float* __restrict__ ed)
{
  const int bn   = blockIdx.x;                // 0..6143
  const int h    = threadIdx.x >> 5;          // 0..3
  const int lane = threadIdx.x & 31;
  const float* w = wh + (long)bn * 1024 + h * 256;
  float s = 0.f, d = 0.f;
  for (int k = lane; k < 256; k += 32) {
    const float v = w[k];
    s += v * asv[h * 256 + k];
    d += v * adv[h * 256 + k];
  }
#pragma unroll
  for (int o = 16; o > 0; o >>= 1) {
    s += __shfl_down(s, o, 32);
    d += __shfl_down(d, o, 32);
  }
  if (lane == 0) {
    const int b = bn / 24, n = bn % 24;
    es[(b * 4 + h) * 24 + n] = s;
    ed[(b * 4 + h) * 24 + n] = d;
  }
}

// ---------------------------------------------------------------------------
// GAT attention softmax + aggregation + ELU.  One block per (b, i).
// ---------------------------------------------------------------------------
__global__ void gat_att(const float* __restrict__ wh, const float* __restrict__ es,
                        const float* __restrict__ ed, const int* __restrict__ adj,
                        const int* __restrict__ mask, float* __restrict__ outp)
{
  const int b = blockIdx.x / 24;
  const int i = blockIdx.x % 24;
  __shared__ float att[4][24];
  const int tid = threadIdx.x;
  if (tid < 96) {
    const int h = tid / 24, j = tid % 24;
    float e = es[(b * 4 + h) * 24 + i] + ed[(b * 4 + h) * 24 + j];
    e = e > 0.f ? e : 0.2f * e;               // leaky_relu(0.2)
    const bool am = ((adj[((long)b * 24 + i) * 24 + j] > 0) &&
                     (mask[b * 24 + i] > 0) && (mask[b * 24 + j] > 0)) || (i == j);
    att[h][j] = am ? e : -1.0e9f;
  }
  __syncthreads();
  if (tid < 4) {
    float mx = -3.0e38f;
    for (int j = 0; j < 24; ++j) mx = fmaxf(mx, att[tid][j]);
    float sm = 0.f;
    for (int j = 0; j < 24; ++j) { const float v = __expf(att[tid][j] - mx); att[tid][j] = v; sm += v; }
    const float inv = 1.f / sm;
    for (int j = 0; j < 24; ++j) att[tid][j] *= inv;
  }
  __syncthreads();
  for (int c = tid; c < 1024; c += 256) {
    const int h = c >> 8;
    float a = 0.f;
    for (int j = 0; j < 24; ++j) a += att[h][j] * wh[((long)(b * 24 + j)) * 1024 + c];
    a = a > 0.f ? a : (__expf(a) - 1.f);      // elu
    outp[((long)(b * 24 + i)) * 1024 + c] = a;
  }
}

// tgt_inter -> encf[:,25600:26624]; masked mean -> encf[:,26624:27648]
__global__ void gat_pool(const float* __restrict__ h3, const int* __restrict__ mask,
                         float* __restrict__ encf)
{
  const int b = blockIdx.x;
  for (int c = threadIdx.x; c < 1024; c += blockDim.x) {
    float sum = 0.f, wsum = 0.f;
    for (int n = 0; n < 24; ++n) {
      const float w = mask[b * 24 + n] > 0 ? 1.f : 0.f;
      sum += w * h3[((long)(b * 24 + n)) * 1024 + c];
      wsum += w;
    }
    encf[(long)b * 27648 + 25600 + c] = h3[((long)(b * 24)) * 1024 + c];
    encf[(long)b * 27648 + 26624 + c] = sum / wsum;
  }
}

__global__ void enc_scale(float* __restrict__ encf, const float* __restrict__ g,
                          const float* __restrict__ bb)
{
  const long i = (long)blockIdx.x * blockDim.x + threadIdx.x;
  const int  c = (int)(i % 27648);
  encf[i] = encf[i] * g[c] + bb[c];
}

__global__ void softmax_out(const float* __restrict__ llat, const float* __restrict__ llon,
                            float* __restrict__ outp)
{
  const int b = threadIdx.x;                  // 256 threads, 1 block
  {
    const float v0 = llat[b * 3], v1 = llat[b * 3 + 1], v2 = llat[b * 3 + 2];
    const float m = fmaxf(v0, fmaxf(v1, v2));
    const float e0 = __expf(v0 - m), e1 = __expf(v1 - m), e2 = __expf(v2 - m);
    const float inv = 1.f / (e0 + e1 + e2);
    outp[b * 3 + 0] = e0 * inv; outp[b * 3 + 1] = e1 * inv; outp[b * 3 + 2] = e2 * inv;
  }
  {
    const float v0 = llon[b * 4], v1 = llon[b * 4 + 1];
    const float v2 = llon[b * 4 + 2], v3 = llon[b * 4 + 3];
    const float m = fmaxf(fmaxf(v0, v1), fmaxf(v2, v3));
    const float e0 = __expf(v0 - m), e1 = __expf(v1 - m);
    const float e2 = __expf(v2 - m), e3 = __expf(v3 - m);
    const float inv = 1.f / (e0 + e1 + e2 + e3);
    outp[768 + b * 4 + 0] = e0 * inv; outp[768 + b * 4 + 1] = e1 * inv;
    outp[768 + b * 4 + 2] = e2 * inv; outp[768 + b * 4 + 3] = e3 * inv;
  }
}

// ---------------------------------------------------------------------------
extern "C" void kernel_launch(void* const* d_in, const int* in_sizes, int n_in,
                              void* d_out, int out_size, void* d_ws, size_t ws_size,
                              hipStream_t stream)
{
  (void)in_sizes; (void)n_in; (void)out_size; (void)ws_size;

  const int*   adj   = (const int*)d_in[0];
  const int*   nmask = (const int*)d_in[1];
  const float* nodes = (const float*)d_in[2];
  const float* b_dec = (const float*)d_in[3];
  const float* b_dev = (const float*)d_in[4];
  const float* b_geo = (const float*)d_in[5];
  const float* b_sur = (const float*)d_in[6];
  const float* b_trj = (const float*)d_in[7];
  const float* bl1b_Wx = (const float*)d_in[8],  *bl1b_Wh = (const float*)d_in[9],  *bl1b_b = (const float*)d_in[10];
  const float* bl1f_Wx = (const float*)d_in[11], *bl1f_Wh = (const float*)d_in[12], *bl1f_b = (const float*)d_in[13];
  const float* bl2b_Wx = (const float*)d_in[14], *bl2b_Wh = (const float*)d_in[15], *bl2b_b = (const float*)d_in[16];
  const float* bl2f_Wx = (const float*)d_in[17], *bl2f_Wh = (const float*)d_in[18], *bl2f_b = (const float*)d_in[19];
  // decoder param sets (lat: 20..28, lon: 29..37), leaf order: Wd0,Wd1,Wo,Wout,Wv,bd0,bd1,bo,bout
  const float* dWd0[2]  = { (const float*)d_in[20], (const float*)d_in[29] };
  const float* dWd1[2]  = { (const float*)d_in[21], (const float*)d_in[30] };
  const float* dWo[2]   = { (const float*)d_in[22], (const float*)d_in[31] };
  const float* dWout[2] = { (const float*)d_in[23], (const float*)d_in[32] };
  const float* dWv[2]   = { (const float*)d_in[24], (const float*)d_in[33] };
  const float* dbd0[2]  = { (const float*)d_in[25], (const float*)d_in[34] };
  const float* dbd1[2]  = { (const float*)d_in[26], (const float*)d_in[35] };
  const float* dbo[2]   = { (const float*)d_in[27], (const float*)d_in[36] };
  const float* dbout[2] = { (const float*)d_in[28], (const float*)d_in[37] };
  const float* eDev_Wx = (const float*)d_in[38], *eDev_Wh = (const float*)d_in[39], *eDev_b = (const float*)d_in[40];
  const float* eGeo_Wx = (const float*)d_in[41], *eGeo_Wh = (const float*)d_in[42], *eGeo_b = (const float*)d_in[43];
  const float* eSur_Wx = (const float*)d_in[44], *eSur_Wh = (const float*)d_in[45], *eSur_b = (const float*)d_in[46];
  const float* eTrj_Wx = (const float*)d_in[47], *eTrj_Wh = (const float*)d_in[48], *eTrj_b = (const float*)d_in[49];
  const float* g_dec = (const float*)d_in[50];
  const float* g_dev = (const float*)d_in[51];
  const float* g_geo = (const float*)d_in[52];
  const float* g_sur = (const float*)d_in[53];
  const float* g_trj = (const float*)d_in[54];
  const float* gat1W  = (const float*)d_in[55];
  const float* gat1ad = (const float*)d_in[56];
  const float* gat1as = (const float*)d_in[57];
  const float* gat2W  = (const float*)d_in[58];
  const float* gat2ad = (const float*)d_in[59];
  const float* gat2as = (const float*)d_in[60];
  const float* x_dev = (const float*)d_in[61];
  const float* x_geo = (const float*)d_in[62];
  const float* x_trj = (const float*)d_in[63];

  // ---- workspace carve (floats), big buffer aliased across phases ----
  float* ws = (float*)d_ws;
  size_t off = 0;
  auto carve = [&](size_t n) { float* p = ws + off; off += (n + 255) & ~(size_t)255; return p; };
  float* buf0 = carve(19660800);  // hcat (6144,3200) -> wh2 -> xp (12800,1024) -> m1 (256,27648)
  float* buf1 = carve(6291456);   // wh1 (6144,1024) -> h3
  float* buf2 = carve(6291456);   // h2
  float* es   = carve(24576);
  float* ed   = carve(24576);
  float* eenc = carve(2457600);   // (256,50,192)
  float* bl1  = carve(6553600);   // (256,50,512)
  float* encf = carve(7077888);   // (256,27648)
  float* m0   = carve(2048);      // (256,8)
  float* m2   = carve(131072);    // (256,512)
  float* m3   = carve(65536);     // (256,256)
  float* llat = carve(1024);
  float* llon = carve(1024);

  auto gemm = [&](const float* A, const float* Bm, const float* bias, float* Cc,
                  int M, int N, int K, int lda, int ldb, int ldc, int act) {
    dim3 g((N + 127) / 128, (M + 15) / 16);
    gemm_f32_wmma<<<g, 256, 0, stream>>>(A, Bm, bias, Cc, M, N, K, lda, ldb, ldc, act);
  };

  // 1) node embedding LSTMs -> hcat (b,node,t,64) == (6144, 3200)
  emb_lstm64<<<16, 128, 0, stream>>>(nodes, g_trj, b_trj, eTrj_Wx, eTrj_Wh, eTrj_b,
                                     buf0, 256, 24, 1, 0, 7, 24, 64, 0);
  emb_lstm64<<<368, 128, 0, stream>>>(nodes, g_sur, b_sur, eSur_Wx, eSur_Wh, eSur_b,
                                      buf0, 5888, 24, 23, 1, 7, 24, 64, 0);

  // 2) GAT layer 1
  gemm(buf0, gat1W, nullptr, buf1, 6144, 1024, 3200, 3200, 1024, 1024, 0);
  gat_esed<<<6144, 128, 0, stream>>>(buf1, gat1as, gat1ad, es, ed);
  gat_att<<<6144, 256, 0, stream>>>(buf1, es, ed, adj, nmask, buf2);

  // 3) GAT layer 2 (wh2 reuses buf0; h3 reuses buf1)
  gemm(buf2, gat2W, nullptr, buf0, 6144, 1024, 1024, 1024, 1024, 1024, 0);
  gat_esed<<<6144, 128, 0, stream>>>(buf0, gat2as, gat2ad, es, ed);
  gat_att<<<6144, 256, 0, stream>>>(buf0, es, ed, adj, nmask, buf1);

  // 4) tgt_inter + masked mean -> tail of encf
  gat_pool<<<256, 256, 0, stream>>>(buf1, nmask, encf);

  // 5) encoder input LSTMs -> eenc (256,50,192) = [traj | geo | dev]
  emb_lstm64<<<16, 128, 0, stream>>>(x_trj, g_trj, b_trj, eTrj_Wx, eTrj_Wh, eTrj_b,
                                     eenc, 256, 1, 1, 0, 7, 1, 192, 0);
  emb_lstm64<<<16, 128, 0, stream>>>(x_geo, g_geo, b_geo, eGeo_Wx, eGeo_Wh, eGeo_b,
                                     eenc, 256, 1, 1, 0, 4, 1, 192, 64);
  emb_lstm64<<<16, 128, 0, stream>>>(x_dev, g_dev, b_dev, eDev_Wx, eDev_Wh, eDev_b,
                                     eenc, 256, 1, 1, 0, 2, 1, 192, 128);

  // 6) BiLSTM 1 -> bl1 (256,50,512)  [xp reuses buf0]
  gemm(eenc, bl1f_Wx, bl1f_b, buf0, 12800, 1024, 192, 192, 1024, 1024, 0);
  big_lstm256<<<16, 256, 0, stream>>>(buf0, bl1f_Wh, bl1, +1, 25600, 512, 0);
  gemm(eenc, bl1b_Wx, bl1b_b, buf0, 12800, 1024, 192, 192, 1024, 1024, 0);
  big_lstm256<<<16, 256, 0, stream>>>(buf0, bl1b_Wh, bl1, -1, 25600, 512, 256);

  // 7) BiLSTM 2 -> encf[:, 0:25600] (row stride 27648)
  gemm(bl1, bl2f_Wx, bl2f_b, buf0, 12800, 1024, 512, 512, 1024, 1024, 0);
  big_lstm256<<<16, 256, 0, stream>>>(buf0, bl2f_Wh, encf, +1, 27648, 512, 0);
  gemm(bl1, bl2b_Wx, bl2b_b, buf0, 12800, 1024, 512, 512, 1024, 1024, 0);
  big_lstm256<<<16, 256, 0, stream>>>(buf0, bl2b_Wh, encf, -1, 27648, 512, 256);

  // 8) enc = enc * g_dec + b_dec
  enc_scale<<<27648, 256, 0, stream>>>(encf, g_dec, b_dec);

  // 9) decoders (m1 reuses buf0)
  float* logit[2] = { llat, llon };
  const int nout[2] = { 3, 4 };
  for (int dsel = 0; dsel < 2; ++dsel) {
    gemm(encf, dWv[dsel], nullptr, m0, 256, 8, 27648, 27648, 8, 8, 0);
    gemm(m0, dWo[dsel], dbo[dsel], buf0, 256, 27648, 8, 8, 27648, 27648, 0);
    gemm(buf0, dWd0[dsel], dbd0[dsel], m2, 256, 512, 27648, 27648, 512, 512, 1);
    gemm(m2, dWd1[dsel], dbd1[dsel], m3, 256, 256, 512, 512, 256, 256, 1);
    gemm(m3, dWout[dsel], dbout[dsel], logit[dsel], 256, nout[dsel], 256, 256, nout[dsel], nout[dsel], 0);
  }

  // 10) softmax heads -> d_out
  softmax_out<<<1, 256, 0, stream>>>(llat, llon, (float*)d_out);
}